// Robust_Model_25795573580202
// MI455X (gfx1250) — compile-verified
//
#include <hip/hip_runtime.h>
#include <hip/hip_bf16.h>

// GCN: h = x@W1+b1; 30x symmetric-normalized propagation; relu; out = h@Wc+bc
// Strategy: build CSR-by-dst once (deg -> scan -> fill), then each hop is a
// pure gather (no atomics in the hot loop). GEMMs use V_WMMA_F32_16X16X4_F32.

#define N_NODES 50000
#define N_EDGES 1600000
#define FEAT_IN 256
#define FEAT_H  128
#define N_CLS   10
#define HOPS    30

typedef __attribute__((ext_vector_type(2))) float v2f;
typedef __attribute__((ext_vector_type(8))) float v8f;

// ---------------------------------------------------------------- degree count
__global__ void gcn_deg_kernel(const int* __restrict__ dst, int* __restrict__ deg) {
    int e = blockIdx.x * blockDim.x + threadIdx.x;
    if (e < N_EDGES) atomicAdd(&deg[dst[e]], 1);
}

// ------------------------------------------------- single-block exclusive scan
// Produces row_ptr[0..N], cursor[i]=row_ptr[i], dinv[i]=rsqrt(deg+1) (self-loop)
__global__ void gcn_scan_kernel(const int* __restrict__ deg,
                                int* __restrict__ row_ptr,
                                int* __restrict__ cursor,
                                float* __restrict__ dinv) {
    __shared__ int lds[1024];
    __shared__ int s_carry;
    if (threadIdx.x == 0) s_carry = 0;
    __syncthreads();
    for (int base = 0; base < N_NODES; base += 1024) {
        int i = base + (int)threadIdx.x;
        int v = (i < N_NODES) ? deg[i] : 0;
        lds[threadIdx.x] = v;
        __syncthreads();
        for (int off = 1; off < 1024; off <<= 1) {
            int t = ((int)threadIdx.x >= off) ? lds[threadIdx.x - off] : 0;
            __syncthreads();
            lds[threadIdx.x] += t;
            __syncthreads();
        }
        int incl = lds[threadIdx.x];
        int c = s_carry;            // read before carry update
        if (i < N_NODES) {
            int rp = c + incl - v;  // exclusive prefix
            row_ptr[i] = rp;
            cursor[i]  = rp;
            dinv[i]    = rsqrtf((float)(v + 1));   // +1 self-loop; deg>=1 always
        }
        __syncthreads();
        if (threadIdx.x == 1023) s_carry = c + lds[1023];
        __syncthreads();
    }
    if (threadIdx.x == 0) row_ptr[N_NODES] = s_carry;
}

// ------------------------------------------------------------------- CSR fill
// csr[k] = { src_node, bitcast(dinv[src]*dinv[dst]) }
__global__ void gcn_fill_kernel(const int* __restrict__ src, const int* __restrict__ dst,
                                int* __restrict__ cursor, const float* __restrict__ dinv,
                                int2* __restrict__ csr) {
    int e = blockIdx.x * blockDim.x + threadIdx.x;
    if (e >= N_EDGES) return;
    int s = src[e], d = dst[e];
    int pos = atomicAdd(&cursor[d], 1);
    int2 ent;
    ent.x = s;
    ent.y = __float_as_int(dinv[s] * dinv[d]);
    csr[pos] = ent;
}

// ------------------------------------------- GEMM1: H = X(50000x256)@W1 + b1
// one wave per 16x16 tile of H; block = 8 waves = one 16-row stripe (128 cols)
__global__ void gcn_gemm1_wmma(const float* __restrict__ X, const float* __restrict__ W,
                               const float* __restrict__ bias, float* __restrict__ H) {
    const int wave = threadIdx.x >> 5;
    const int lane = threadIdx.x & 31;
    const int m0 = blockIdx.x * 16;       // 3125 blocks, exact
    const int n0 = wave * 16;             // 8 waves cover 128 cols
    const int li = lane & 15;
    const int kh = (lane >> 4) << 1;      // lanes 0-15: K+{0,1}; 16-31: K+{2,3}
    v8f acc = {};
    const float* xrow = X + (size_t)(m0 + li) * FEAT_IN;
    for (int k = 0; k < FEAT_IN; k += 4) {
        v2f a = *(const v2f*)(xrow + k + kh);          // x[m][k+kh], x[m][k+kh+1]
        v2f b;
        b.x = W[(size_t)(k + kh)     * FEAT_H + n0 + li];
        b.y = W[(size_t)(k + kh + 1) * FEAT_H + n0 + li];
        acc = __builtin_amdgcn_wmma_f32_16x16x4_f32(false, a, false, b,
                                                    (short)0, acc, false, false);
    }
    const float bv = bias[n0 + li];
    const int mbase = (lane >> 4) * 8;
#pragma unroll
    for (int g = 0; g < 8; ++g)
        H[(size_t)(m0 + mbase + g) * FEAT_H + n0 + li] = acc[g] + bv;
}

// -------------------------------------------------- propagation hop (gather)
// one wave per node; lane owns one float4 of the 128-wide row
__global__ void gcn_hop_kernel(const float* __restrict__ hin, float* __restrict__ hout,
                               const int* __restrict__ row_ptr,
                               const int2* __restrict__ csr,
                               const float* __restrict__ dinv) {
    const int node = (int)((blockIdx.x * blockDim.x + threadIdx.x) >> 5);
    const int lane = threadIdx.x & 31;
    if (node >= N_NODES) return;                 // grid exact; wave-uniform
    const float4* __restrict__ hin4 = (const float4*)hin;
    float4* __restrict__ hout4 = (float4*)hout;
    const float d  = dinv[node];
    const float dd = d * d;
    float4 self = hin4[(size_t)node * 32 + lane];
    float4 acc;
    acc.x = self.x * dd; acc.y = self.y * dd;
    acc.z = self.z * dd; acc.w = self.w * dd;
    int k   = row_ptr[node];
    int end = row_ptr[node + 1];
    for (; k < end; ++k) {
        int2 ent = csr[k];                       // broadcast (same addr all lanes)
        float w  = __int_as_float(ent.y);
        float4 v = hin4[(size_t)ent.x * 32 + lane];  // 512B contiguous per wave
        acc.x = fmaf(w, v.x, acc.x);
        acc.y = fmaf(w, v.y, acc.y);
        acc.z = fmaf(w, v.z, acc.z);
        acc.w = fmaf(w, v.w, acc.w);
    }
    hout4[(size_t)node * 32 + lane] = acc;
}

// ------------------------------------- classifier: OUT = relu(H)@Wc + bc (N=10)
__global__ void gcn_gemm2_wmma(const float* __restrict__ H, const float* __restrict__ Wc,
                               const float* __restrict__ bc, float* __restrict__ OUT) {
    const int gwave = (int)((blockIdx.x * blockDim.x + threadIdx.x) >> 5);
    const int lane = threadIdx.x & 31;
    if (gwave >= N_NODES / 16) return;           // wave-uniform exit
    const int m0 = gwave * 16;
    const int li = lane & 15;
    const int kh = (lane >> 4) << 1;
    const bool nvalid = li < N_CLS;              // pad N=10 -> 16 with zeros
    v8f acc = {};
    const float* hrow = H + (size_t)(m0 + li) * FEAT_H;
    for (int k = 0; k < FEAT_H; k += 4) {
        v2f a;
        a.x = fmaxf(hrow[k + kh], 0.0f);         // fused ReLU
        a.y = fmaxf(hrow[k + kh + 1], 0.0f);
        v2f b;
        b.x = nvalid ? Wc[(k + kh)     * N_CLS + li] : 0.0f;
        b.y = nvalid ? Wc[(k + kh + 1) * N_CLS + li] : 0.0f;
        acc = __builtin_amdgcn_wmma_f32_16x16x4_f32(false, a, false, b,
                                                    (short)0, acc, false, false);
    }
    if (nvalid) {
        const float bv = bc[li];
        const int mbase = (lane >> 4) * 8;
#pragma unroll
        for (int g = 0; g < 8; ++g)
            OUT[(size_t)(m0 + mbase + g) * N_CLS + li] = acc[g] + bv;
    }
}

extern "C" void kernel_launch(void* const* d_in, const int* in_sizes, int n_in,
                              void* d_out, int out_size, void* d_ws, size_t ws_size,
                              hipStream_t stream) {
    const float* x  = (const float*)d_in[0];
    const int*   ei = (const int*)d_in[1];     // [2, E] int32
    const float* W1 = (const float*)d_in[2];
    const float* b1 = (const float*)d_in[3];
    const float* Wc = (const float*)d_in[4];
    const float* bc = (const float*)d_in[5];
    float* out = (float*)d_out;
    const int* src = ei;
    const int* dst = ei + N_EDGES;

    // workspace carve-out (256B aligned slices)
    char* ws = (char*)d_ws;
    size_t off = 0;
    auto carve = [&](size_t bytes) -> void* {
        void* p = ws + off;
        off += (bytes + 255) & ~(size_t)255;
        return p;
    };
    int*   deg     = (int*)  carve((size_t)N_NODES * 4);
    int*   row_ptr = (int*)  carve((size_t)(N_NODES + 1) * 4);
    int*   cursor  = (int*)  carve((size_t)N_NODES * 4);
    float* dinv    = (float*)carve((size_t)N_NODES * 4);
    int2*  csr     = (int2*) carve((size_t)N_EDGES * 8);
    float* hA      = (float*)carve((size_t)N_NODES * FEAT_H * 4);
    float* hB      = (float*)carve((size_t)N_NODES * FEAT_H * 4);

    // 1) degrees (must re-zero each call)
    hipMemsetAsync(deg, 0, (size_t)N_NODES * 4, stream);
    gcn_deg_kernel<<<(N_EDGES + 255) / 256, 256, 0, stream>>>(dst, deg);
    // 2) scan -> row_ptr/cursor/dinv
    gcn_scan_kernel<<<1, 1024, 0, stream>>>(deg, row_ptr, cursor, dinv);
    // 3) CSR fill
    gcn_fill_kernel<<<(N_EDGES + 255) / 256, 256, 0, stream>>>(src, dst, cursor, dinv, csr);
    // 4) input GEMM (WMMA f32)
    gcn_gemm1_wmma<<<N_NODES / 16, 256, 0, stream>>>(x, W1, b1, hA);
    // 5) 30 propagation hops (gather; h is L2-resident at 25.6MB)
    float* cur = hA;
    float* nxt = hB;
    for (int t = 0; t < HOPS; ++t) {
        gcn_hop_kernel<<<N_NODES / 8, 256, 0, stream>>>(cur, nxt, row_ptr, csr, dinv);
        float* tmp = cur; cur = nxt; nxt = tmp;
    }
    // 6) ReLU + classifier GEMM (WMMA f32, N padded 10->16)
    const int waves2 = N_NODES / 16;               // 3125
    gcn_gemm2_wmma<<<(waves2 + 7) / 8, 256, 0, stream>>>(cur, Wc, bc, out);
}